// CentroidInstanceLoss_24764781428790
// MI455X (gfx1250) — compile-verified
//
#include <hip/hip_runtime.h>
#include <hip/hip_bf16.h>
#include <math.h>

// CentroidInstanceLoss for MI455X (gfx1250, wave32).
// Memory-bound: 2 streaming passes over 128MB of f32 -> ~12.5us floor @23.3TB/s.
// f32 end to end; WMMA f32 16x16x4 used as an exact cross-lane reduction.

typedef __attribute__((ext_vector_type(2))) float v2f;
typedef __attribute__((ext_vector_type(8))) float v8f;

#define NSEG 512      // B(8) * L(64) segments
#define SEGP 17       // padded row stride (16 sums + 1 count); 17 coprime 64 -> no LDS bank conflicts
#define NLAB 64
#define NSB  8

// ---- wave32 sum via V_WMMA_F32_16X16X4_F32 (exact f32) -------------------
// A layout (16x4 f32): VGPR0: lanes0-15 -> (M=lane,K=0), lanes16-31 -> (M=lane-16,K=2)
//                      VGPR1: lanes0-15 -> K=1,          lanes16-31 -> K=3
// With A.y=0 and B=ones: D[m,n] = p_m + p_{m+16} (broadcast over n).
// D layout: VGPR v, lanes0-15 = D[v,*], lanes16-31 = D[v+8,*].
// Sum of a lane's 8 D regs = half-total; shfl_xor(16) completes the wave sum.
__device__ __forceinline__ float wave_sum_wmma(float v) {
  v2f a; a.x = v;    a.y = 0.0f;
  v2f b; b.x = 1.0f; b.y = 1.0f;
  v8f c = {};
  c = __builtin_amdgcn_wmma_f32_16x16x4_f32(false, a, false, b, (short)0, c, false, false);
  float s = c[0] + c[1] + c[2] + c[3] + c[4] + c[5] + c[6] + c[7];
  s += __shfl_xor(s, 16, 32);
  return s;  // full 32-lane sum in every lane
}

__device__ __forceinline__ float block_sum(float v, float* redbuf) {
  // All threads must reach this (EXEC all-ones for WMMA): call in uniform flow.
  float s = wave_sum_wmma(v);
  int lane = threadIdx.x & 31;
  int wv   = threadIdx.x >> 5;
  __syncthreads();
  if (lane == 0) redbuf[wv] = s;
  __syncthreads();
  float t = 0.0f;
  if (threadIdx.x == 0) {
    int nw = (int)(blockDim.x >> 5);
    for (int w = 0; w < nw; ++w) t += redbuf[w];
  }
  return t;  // valid on thread 0 only
}

// ---- kernel 0: zero the global accumulators and the output ----------------
__global__ __launch_bounds__(256) void k_init(float* __restrict__ gacc,
                                              float* __restrict__ out, int out_n) {
  int i = blockIdx.x * blockDim.x + threadIdx.x;
  int total = NSEG * SEGP + out_n;
  if (i < NSEG * SEGP) gacc[i] = 0.0f;
  else if (i < total)  out[i - NSEG * SEGP] = 0.0f;
}

// ---- kernel 1: normalize + segment sums/counts (LDS accumulate) -----------
__global__ __launch_bounds__(256) void k_segacc(const float* __restrict__ x,
                                                const int* __restrict__ lab,
                                                const int* __restrict__ sb,
                                                float* __restrict__ gacc, int n) {
  __shared__ float lacc[NSEG * SEGP];  // 34816 B
  for (int i = threadIdx.x; i < NSEG * SEGP; i += blockDim.x) lacc[i] = 0.0f;
  __syncthreads();

  const int stride = gridDim.x * blockDim.x;
  const float4* __restrict__ xr = reinterpret_cast<const float4*>(x);
  for (int i = blockIdx.x * blockDim.x + threadIdx.x; i < n; i += stride) {
    // prefetch next grid-stride iteration into L2 (16MB/iter << 192MB L2)
    __builtin_prefetch(xr + (size_t)4 * (i + stride), 0, 0);
    float4 a = xr[(size_t)4 * i + 0];
    float4 b = xr[(size_t)4 * i + 1];
    float4 c = xr[(size_t)4 * i + 2];
    float4 d = xr[(size_t)4 * i + 3];
    float ss = a.x*a.x + a.y*a.y + a.z*a.z + a.w*a.w
             + b.x*b.x + b.y*b.y + b.z*b.z + b.w*b.w
             + c.x*c.x + c.y*c.y + c.z*c.z + c.w*c.w
             + d.x*d.x + d.y*d.y + d.z*d.z + d.w*d.w;
    float sc = 1.0f / (sqrtf(ss) + 1e-8f);
    int seg = sb[i] * NLAB + lab[i];
    float* q = lacc + seg * SEGP;   // 17-stride: conflict-free across a wave
    atomicAdd(q + 0,  a.x * sc); atomicAdd(q + 1,  a.y * sc);
    atomicAdd(q + 2,  a.z * sc); atomicAdd(q + 3,  a.w * sc);
    atomicAdd(q + 4,  b.x * sc); atomicAdd(q + 5,  b.y * sc);
    atomicAdd(q + 6,  b.z * sc); atomicAdd(q + 7,  b.w * sc);
    atomicAdd(q + 8,  c.x * sc); atomicAdd(q + 9,  c.y * sc);
    atomicAdd(q + 10, c.z * sc); atomicAdd(q + 11, c.w * sc);
    atomicAdd(q + 12, d.x * sc); atomicAdd(q + 13, d.y * sc);
    atomicAdd(q + 14, d.z * sc); atomicAdd(q + 15, d.w * sc);
    atomicAdd(q + 16, 1.0f);
  }
  __syncthreads();
  for (int idx = threadIdx.x; idx < NSEG * SEGP; idx += blockDim.x)
    atomicAdd(gacc + idx, lacc[idx]);  // global layout mirrors padded LDS layout
}

// ---- kernel 2: centroids + per-segment pull scale + push term -------------
__global__ __launch_bounds__(256) void k_centroid_push(const float* __restrict__ gacc,
                                                       float* __restrict__ gmu,
                                                       float* __restrict__ gps,
                                                       float* __restrict__ out) {
  __shared__ float mu[NLAB * SEGP];  // this subbatch's 64 centroids, padded
  __shared__ float red[8];
  const int b = blockIdx.x;
  for (int idx = threadIdx.x; idx < NLAB * 16; idx += blockDim.x) {
    int l = idx >> 4, d = idx & 15;
    int s = b * NLAB + l;
    float cnt = gacc[s * SEGP + 16];
    float m   = gacc[s * SEGP + d] / cnt;
    mu[l * SEGP + d] = m;
    gmu[s * 16 + d]  = m;
    if (d == 0) gps[s] = 1.0f / (64.0f * cnt * 8.0f);  // 1/(M*Nm*B)
  }
  __syncthreads();

  float acc = 0.0f;
  for (int p = threadIdx.x; p < NLAB * NLAB; p += blockDim.x) {
    int i = p >> 6, j = p & 63;
    if (i != j) {
      const float* mi = mu + i * SEGP;
      const float* mj = mu + j * SEGP;
      float dist = 0.0f;
#pragma unroll
      for (int d = 0; d < 16; ++d) dist += fabsf(mi[d] - mj[d]);
      float h = fmaxf(0.0f, 3.0f - dist);  // 2*DELTA_D = 3.0
      acc += h * h;
    }
  }
  float tot = block_sum(acc, red);
  if (threadIdx.x == 0)
    atomicAdd(out, tot * (1.0f / (64.0f * 63.0f * 8.0f)));  // /(M*(M-1))/B
}

// ---- kernel 3: pull pass (centroid table staged in LDS) -------------------
__global__ __launch_bounds__(256) void k_pull(const float* __restrict__ x,
                                              const int* __restrict__ lab,
                                              const int* __restrict__ sb,
                                              const float* __restrict__ gmu,
                                              const float* __restrict__ gps,
                                              float* __restrict__ out, int n) {
  __shared__ float mu[NSEG * SEGP];  // 34816 B, padded stride 17
  __shared__ float psc[NSEG];
  __shared__ float red[8];
  for (int idx = threadIdx.x; idx < NSEG * 16; idx += blockDim.x) {
    int s = idx >> 4, d = idx & 15;
    mu[s * SEGP + d] = gmu[idx];
  }
  for (int idx = threadIdx.x; idx < NSEG; idx += blockDim.x) psc[idx] = gps[idx];
  __syncthreads();

  const int stride = gridDim.x * blockDim.x;
  const float4* __restrict__ xr = reinterpret_cast<const float4*>(x);
  float acc = 0.0f;
  for (int i = blockIdx.x * blockDim.x + threadIdx.x; i < n; i += stride) {
    __builtin_prefetch(xr + (size_t)4 * (i + stride), 0, 0);
    float4 a = xr[(size_t)4 * i + 0];
    float4 b = xr[(size_t)4 * i + 1];
    float4 c = xr[(size_t)4 * i + 2];
    float4 d = xr[(size_t)4 * i + 3];
    float ss = a.x*a.x + a.y*a.y + a.z*a.z + a.w*a.w
             + b.x*b.x + b.y*b.y + b.z*b.z + b.w*b.w
             + c.x*c.x + c.y*c.y + c.z*c.z + c.w*c.w
             + d.x*d.x + d.y*d.y + d.z*d.z + d.w*d.w;
    float scl = 1.0f / (sqrtf(ss) + 1e-8f);
    int seg = sb[i] * NLAB + lab[i];
    const float* m = mu + seg * SEGP;
    float d1 = fabsf(m[0]  - a.x*scl) + fabsf(m[1]  - a.y*scl)
             + fabsf(m[2]  - a.z*scl) + fabsf(m[3]  - a.w*scl)
             + fabsf(m[4]  - b.x*scl) + fabsf(m[5]  - b.y*scl)
             + fabsf(m[6]  - b.z*scl) + fabsf(m[7]  - b.w*scl)
             + fabsf(m[8]  - c.x*scl) + fabsf(m[9]  - c.y*scl)
             + fabsf(m[10] - c.z*scl) + fabsf(m[11] - c.w*scl)
             + fabsf(m[12] - d.x*scl) + fabsf(m[13] - d.y*scl)
             + fabsf(m[14] - d.z*scl) + fabsf(m[15] - d.w*scl);
    float h = fmaxf(0.0f, d1 - 0.5f);  // DELTA_V
    acc += h * h * psc[seg];
  }
  float tot = block_sum(acc, red);
  if (threadIdx.x == 0) atomicAdd(out, tot);
}

extern "C" void kernel_launch(void* const* d_in, const int* in_sizes, int n_in,
                              void* d_out, int out_size, void* d_ws, size_t ws_size,
                              hipStream_t stream) {
  const float* x  = (const float*)d_in[0];
  const int* lab  = (const int*)d_in[1];
  const int* sbi  = (const int*)d_in[2];
  float* out      = (float*)d_out;
  const int n     = in_sizes[0] / 16;  // points

  float* gacc = (float*)d_ws;          // [512*17] sums+count, padded
  float* gmu  = gacc + NSEG * SEGP;    // [512*16] centroids
  float* gps  = gmu + NSEG * 16;       // [512]    per-segment pull scale

  const int blk = 256;
  const int init_elems = NSEG * SEGP + out_size;
  k_init<<<(init_elems + blk - 1) / blk, blk, 0, stream>>>(gacc, out, out_size);
  k_segacc<<<1024, blk, 0, stream>>>(x, lab, sbi, gacc, n);
  k_centroid_push<<<NSB, blk, 0, stream>>>(gacc, gmu, gps, out);
  k_pull<<<1024, blk, 0, stream>>>(x, lab, sbi, gmu, gps, out, n);
}